// FullHANTeacher_39633958208186
// MI455X (gfx1250) — compile-verified
//
#include <hip/hip_runtime.h>
#include <hip/hip_bf16.h>

#define N_A 100000
#define N_P 500000
#define HIDDEN 128
#define HEADS 4
#define D_HEAD 32
#define NUM_CLASSES 16
#define E_PA 1500000
#define E_AA 1000000
#define NEG_SLOPE 0.2f

typedef __attribute__((ext_vector_type(2))) float v2f;
typedef __attribute__((ext_vector_type(8))) float v8f;

// ---------- helpers ----------
__device__ __forceinline__ unsigned orderedBits(float f) {
    unsigned u = __float_as_uint(f);
    return (u & 0x80000000u) ? ~u : (u | 0x80000000u);
}
__device__ __forceinline__ float fromOrdered(unsigned u) {
    return __uint_as_float((u & 0x80000000u) ? (u & 0x7fffffffu) : ~u);
}
__device__ __forceinline__ float lrelu(float f) {
    return f >= 0.0f ? f : NEG_SLOPE * f;
}

// async global->LDS b128 transfer (CDNA5 ASYNCcnt path); lds_off = low 32 bits of
// the generic shared-memory address (== workgroup-relative LDS byte offset).
__device__ __forceinline__ void async_g2lds_b128(unsigned lds_off, const float* gptr) {
    asm volatile("global_load_async_to_lds_b128 %0, %1, off"
                 :: "v"(lds_off), "v"(gptr) : "memory");
}
__device__ __forceinline__ void wait_async0() {
#if __has_builtin(__builtin_amdgcn_s_wait_asynccnt)
    __builtin_amdgcn_s_wait_asynccnt(0);
#else
    asm volatile("s_wait_asynccnt 0x0" ::: "memory");
#endif
}

// ---------- zero init of workspace region ----------
__global__ void zero_init_kernel(float* __restrict__ p, size_t n) {
    size_t t = (size_t)blockIdx.x * blockDim.x + threadIdx.x;
    size_t stride = (size_t)gridDim.x * blockDim.x;
    for (size_t i = t; i < n; i += stride) p[i] = 0.0f;
}

// ---------- projection GEMM: Y[N,128] = X[N,128] @ W[128,128]^T + bias ----------
// One block = 16 rows; 8 waves each produce a 16x16 column tile via f32 WMMA.
// Full tiles stage A via async global->LDS DMA (no VGPR round trip).
__global__ __launch_bounds__(256)
void proj_gemm_kernel(const float* __restrict__ X, const float* __restrict__ W,
                      const float* __restrict__ bias, float* __restrict__ Y, int N) {
    __shared__ float ldsA[16 * 128];
    const int tid  = threadIdx.x;
    const int row0 = blockIdx.x * 16;
    if (row0 + 16 <= N) {
        // tile is 2048 contiguous floats = 512 x 16B; 2 async b128 per thread
        const float* gsrc = X + (size_t)row0 * 128;
        const unsigned lds_base = (unsigned)(size_t)ldsA;
        #pragma unroll
        for (int j = 0; j < 2; ++j) {
            int i4 = j * 256 + tid;
            async_g2lds_b128(lds_base + (unsigned)i4 * 16u, gsrc + i4 * 4);
        }
        wait_async0();
    } else {
        for (int j = 0; j < 8; ++j) {
            int i = j * 256 + tid;
            int r = i >> 7, c = i & 127;
            ldsA[i] = (row0 + r < N) ? X[(size_t)(row0 + r) * 128 + c] : 0.0f;
        }
    }
    __syncthreads();
    const int wave = tid >> 5;
    const int lane = tid & 31;
    const int half = lane >> 4;      // K sub-block selector (A: K=2*half.., B: same)
    const int mn   = lane & 15;      // A: row M; B: col N
    const int col0 = wave * 16;
    const float* wrow = W + (size_t)(col0 + mn) * 128 + 2 * half; // B[k][n] = W[n][k]
    const float* arow = ldsA + mn * 128 + 2 * half;
    v8f acc = {};
    for (int kk = 0; kk < 128; kk += 4) {
        v2f a = { arow[kk], arow[kk + 1] };
        v2f b = { wrow[kk], wrow[kk + 1] };
        acc = __builtin_amdgcn_wmma_f32_16x16x4_f32(false, a, false, b,
                                                    (short)0, acc, false, false);
    }
    float bv = bias[col0 + mn];
    #pragma unroll
    for (int r = 0; r < 8; ++r) {
        int m = r + 8 * half;
        if (row0 + m < N)
            Y[(size_t)(row0 + m) * 128 + col0 + mn] = acc[r] + bv;
    }
}

// ---------- per-(node,head) attention coefficients: a[n,h] = <x[n,h,:], lin[h,:]> ----------
__global__ void node_coeffs_kernel(const float* __restrict__ Xh, const float* __restrict__ lin,
                                   float* __restrict__ a, int N) {
    int t = blockIdx.x * blockDim.x + threadIdx.x;
    if (t >= N * HEADS) return;
    int node = t >> 2, h = t & 3;
    const float* x = Xh + (size_t)node * 128 + h * D_HEAD;
    const float* l = lin + h * D_HEAD;
    float s = 0.0f;
    #pragma unroll
    for (int d = 0; d < D_HEAD; ++d) s += x[d] * l[d];
    a[t] = s;
}

// ---------- pass 1: segment-max of leaky-relu'd logits via ordered-int atomicMax ----------
__global__ void edge_max_kernel(const int* __restrict__ src, const int* __restrict__ dst,
                                const float* __restrict__ asrc, const float* __restrict__ adst,
                                unsigned* __restrict__ amax, int E) {
    int t = blockIdx.x * blockDim.x + threadIdx.x;
    if (t >= E * HEADS) return;
    int e = t >> 2, h = t & 3;
    int s = src[e], d = dst[e];
    float al = lrelu(asrc[s * 4 + h] + adst[d * 4 + h]);
    atomicMax(&amax[d * 4 + h], orderedBits(al));
}

// ---------- pass 2: exp + denominator, cache per-edge exp ----------
__global__ void edge_den_kernel(const int* __restrict__ src, const int* __restrict__ dst,
                                const float* __restrict__ asrc, const float* __restrict__ adst,
                                const unsigned* __restrict__ amax, float* __restrict__ denom,
                                float* __restrict__ eexp, int E) {
    int t = blockIdx.x * blockDim.x + threadIdx.x;
    if (t >= E * HEADS) return;
    int e = t >> 2, h = t & 3;
    int s = src[e], d = dst[e];
    float al = lrelu(asrc[s * 4 + h] + adst[d * 4 + h]);
    unsigned mb = amax[d * 4 + h];
    float mx = mb ? fromOrdered(mb) : 0.0f;
    float ev = expf(al - mx);
    eexp[(size_t)e * 4 + h] = ev;
    atomicAdd(&denom[d * 4 + h], ev);
}

// ---------- pass 3: weighted message scatter; one edge per 32-lane wave ----------
// Edge metadata is wave-uniform: force it onto the scalar path so the vector
// pipe only carries the gather + coalesced float atomics.
__global__ __launch_bounds__(256)
void edge_msg_kernel(const int* __restrict__ src, const int* __restrict__ dst,
                     const float* __restrict__ Xsrc, const float* __restrict__ eexp,
                     const float* __restrict__ denom, float* __restrict__ out, int E) {
    int wv = __builtin_amdgcn_readfirstlane((int)(threadIdx.x >> 5));
    int e = blockIdx.x * 8 + wv;
    if (e >= E) return;
    int lane = threadIdx.x & 31;
    int s = __builtin_amdgcn_readfirstlane(src[e]);
    int d = __builtin_amdgcn_readfirstlane(dst[e]);
    const float* xs = Xsrc + (size_t)s * 128;
    float* od = out + (size_t)d * 128;
    __builtin_prefetch(xs, 0, 0);   // global_prefetch_b8 on the gathered source row
    #pragma unroll
    for (int k = 0; k < HEADS; ++k) {  // head == k since dim = k*32 + lane
        float coef = eexp[(size_t)e * 4 + k] / (denom[d * 4 + k] + 1e-16f);
        int dim = k * 32 + lane;
        atomicAdd(od + dim, xs[dim] * coef);
    }
}

// ---------- semantic score: sum_n q . tanh(relu(S[n]) @ Kw^T + kb), via WMMA ----------
__global__ __launch_bounds__(256)
void sem_score_kernel(const float* __restrict__ S, const float* __restrict__ Kw,
                      const float* __restrict__ kb, const float* __restrict__ q,
                      float* __restrict__ score_acc, int N) {
    __shared__ float ldsA[16 * 128];
    __shared__ float red[256];
    const int tid  = threadIdx.x;
    const int row0 = blockIdx.x * 16;
    for (int j = 0; j < 8; ++j) {
        int i = j * 256 + tid;
        int r = i >> 7, c = i & 127;
        float v = 0.0f;
        if (row0 + r < N) {
            v = S[(size_t)(row0 + r) * 128 + c];
            v = v > 0.0f ? v : 0.0f;   // relu fused at read
        }
        ldsA[i] = v;
    }
    __syncthreads();
    const int wave = tid >> 5, lane = tid & 31;
    const int half = lane >> 4, mn = lane & 15;
    const int col0 = wave * 16;
    const float* wrow = Kw + (size_t)(col0 + mn) * 128 + 2 * half;
    const float* arow = ldsA + mn * 128 + 2 * half;
    v8f acc = {};
    for (int kk = 0; kk < 128; kk += 4) {
        v2f a = { arow[kk], arow[kk + 1] };
        v2f b = { wrow[kk], wrow[kk + 1] };
        acc = __builtin_amdgcn_wmma_f32_16x16x4_f32(false, a, false, b,
                                                    (short)0, acc, false, false);
    }
    float qv = q[col0 + mn], bv = kb[col0 + mn];
    float partial = 0.0f;
    #pragma unroll
    for (int r = 0; r < 8; ++r) {
        int m = r + 8 * half;
        if (row0 + m < N) partial += tanhf(acc[r] + bv) * qv;
    }
    red[tid] = partial;
    __syncthreads();
    for (int sft = 128; sft > 0; sft >>= 1) {
        if (tid < sft) red[tid] += red[tid + sft];
        __syncthreads();
    }
    if (tid == 0) atomicAdd(score_acc, red[0]);
}

// ---------- softmax over the 2 metapath scores ----------
__global__ void softmax2_kernel(const float* __restrict__ score_acc, float* __restrict__ attn) {
    if (threadIdx.x == 0) {
        float s0 = score_acc[0] / (float)N_A;
        float s1 = score_acc[1] / (float)N_A;
        float m  = fmaxf(s0, s1);
        float e0 = expf(s0 - m), e1 = expf(s1 - m);
        float inv = 1.0f / (e0 + e1);
        attn[0] = e0 * inv;
        attn[1] = e1 * inv;
    }
}

// ---------- fuse + H_out + alpha_sem + logits (WMMA, 16 classes) ----------
__global__ __launch_bounds__(128)
void fuse_out_kernel(const float* __restrict__ out_pa, const float* __restrict__ out_aa,
                     const float* __restrict__ attn, const float* __restrict__ Ow,
                     const float* __restrict__ Ob, float* __restrict__ logits,
                     float* __restrict__ Hout, float* __restrict__ alpha_sem, int N) {
    __shared__ float ldsh[16 * 128];
    const int tid  = threadIdx.x;
    const int row0 = blockIdx.x * 16;
    float a0 = attn[0], a1 = attn[1];
    for (int j = 0; j < 16; ++j) {
        int i = j * 128 + tid;
        int r = i >> 7, c = i & 127;
        float h = 0.0f;
        if (row0 + r < N) {
            float p  = out_pa[(size_t)(row0 + r) * 128 + c]; p  = p  > 0.0f ? p  : 0.0f;
            float aa = out_aa[(size_t)(row0 + r) * 128 + c]; aa = aa > 0.0f ? aa : 0.0f;
            h = a0 * p + a1 * aa;
            Hout[((size_t)(row0 + r) * 2 + 0) * 128 + c] = h;
            Hout[((size_t)(row0 + r) * 2 + 1) * 128 + c] = h;
        }
        ldsh[i] = h;
    }
    if (tid < 32 && (row0 + (tid >> 1)) < N)
        alpha_sem[(size_t)(row0 + (tid >> 1)) * 2 + (tid & 1)] = 0.5f;
    __syncthreads();
    if (tid < 32) {   // uniform per wave: wave 0 fully active, others branch around
        const int lane = tid;
        const int half = lane >> 4, mn = lane & 15;
        const float* wrow = Ow + (size_t)mn * 128 + 2 * half;  // B[k][cls] = Ow[cls][k]
        const float* arow = ldsh + mn * 128 + 2 * half;
        v8f acc = {};
        for (int kk = 0; kk < 128; kk += 4) {
            v2f a = { arow[kk], arow[kk + 1] };
            v2f b = { wrow[kk], wrow[kk + 1] };
            acc = __builtin_amdgcn_wmma_f32_16x16x4_f32(false, a, false, b,
                                                        (short)0, acc, false, false);
        }
        float bv = Ob[mn];
        #pragma unroll
        for (int r = 0; r < 8; ++r) {
            int m = r + 8 * half;
            if (row0 + m < N)
                logits[(size_t)(row0 + m) * 16 + mn] = acc[r] + bv;
        }
    }
}

extern "C" void kernel_launch(void* const* d_in, const int* in_sizes, int n_in,
                              void* d_out, int out_size, void* d_ws, size_t ws_size,
                              hipStream_t stream) {
    (void)in_sizes; (void)n_in; (void)out_size; (void)ws_size;
    const float* x_author = (const float*)d_in[0];
    const float* x_paper  = (const float*)d_in[1];
    const float* paw = (const float*)d_in[2];
    const float* pab = (const float*)d_in[3];
    const float* ppw = (const float*)d_in[4];
    const float* ppb = (const float*)d_in[5];
    const float* ls_pa = (const float*)d_in[6];
    const float* ld_pa = (const float*)d_in[7];
    const float* ls_aa = (const float*)d_in[8];
    const float* ld_aa = (const float*)d_in[9];
    const float* q  = (const float*)d_in[10];
    const float* kw = (const float*)d_in[11];
    const float* kb = (const float*)d_in[12];
    const float* ow = (const float*)d_in[13];
    const float* ob = (const float*)d_in[14];
    const int* pa_src = (const int*)d_in[15];
    const int* pa_dst = (const int*)d_in[16];
    const int* aa_src = (const int*)d_in[17];
    const int* aa_dst = (const int*)d_in[18];

    float* ws = (float*)d_ws;
    size_t off = 0;
    float* xa = ws + off;       off += (size_t)N_A * 128;
    float* xp = ws + off;       off += (size_t)N_P * 128;
    size_t zero_begin = off;
    float* out_pa = ws + off;   off += (size_t)N_A * 128;
    float* out_aa = ws + off;   off += (size_t)N_A * 128;
    float* a_pa_src = ws + off; off += (size_t)N_P * 4;
    float* a_pa_dst = ws + off; off += (size_t)N_A * 4;
    float* a_aa_src = ws + off; off += (size_t)N_A * 4;
    float* a_aa_dst = ws + off; off += (size_t)N_A * 4;
    unsigned* amax_pa = (unsigned*)(ws + off); off += (size_t)N_A * 4;
    float* denom_pa = ws + off; off += (size_t)N_A * 4;
    unsigned* amax_aa = (unsigned*)(ws + off); off += (size_t)N_A * 4;
    float* denom_aa = ws + off; off += (size_t)N_A * 4;
    float* exp_pa = ws + off;   off += (size_t)E_PA * 4;
    float* exp_aa = ws + off;   off += (size_t)E_AA * 4;
    float* score = ws + off;    off += 2;
    float* attn  = ws + off;    off += 2;
    size_t zero_count = off - zero_begin;

    float* logits    = (float*)d_out;
    float* Hout      = logits + (size_t)N_A * 16;
    float* alpha_sem = Hout + (size_t)N_A * 256;

    zero_init_kernel<<<2048, 256, 0, stream>>>(ws + zero_begin, zero_count);

    proj_gemm_kernel<<<(N_A + 15) / 16, 256, 0, stream>>>(x_author, paw, pab, xa, N_A);
    proj_gemm_kernel<<<(N_P + 15) / 16, 256, 0, stream>>>(x_paper,  ppw, ppb, xp, N_P);

    node_coeffs_kernel<<<(N_P * 4 + 255) / 256, 256, 0, stream>>>(xp, ls_pa, a_pa_src, N_P);
    node_coeffs_kernel<<<(N_A * 4 + 255) / 256, 256, 0, stream>>>(xa, ld_pa, a_pa_dst, N_A);
    node_coeffs_kernel<<<(N_A * 4 + 255) / 256, 256, 0, stream>>>(xa, ls_aa, a_aa_src, N_A);
    node_coeffs_kernel<<<(N_A * 4 + 255) / 256, 256, 0, stream>>>(xa, ld_aa, a_aa_dst, N_A);

    edge_max_kernel<<<(E_PA * 4 + 255) / 256, 256, 0, stream>>>(pa_src, pa_dst, a_pa_src, a_pa_dst, amax_pa, E_PA);
    edge_max_kernel<<<(E_AA * 4 + 255) / 256, 256, 0, stream>>>(aa_src, aa_dst, a_aa_src, a_aa_dst, amax_aa, E_AA);

    edge_den_kernel<<<(E_PA * 4 + 255) / 256, 256, 0, stream>>>(pa_src, pa_dst, a_pa_src, a_pa_dst, amax_pa, denom_pa, exp_pa, E_PA);
    edge_den_kernel<<<(E_AA * 4 + 255) / 256, 256, 0, stream>>>(aa_src, aa_dst, a_aa_src, a_aa_dst, amax_aa, denom_aa, exp_aa, E_AA);

    edge_msg_kernel<<<(E_PA + 7) / 8, 256, 0, stream>>>(pa_src, pa_dst, xp, exp_pa, denom_pa, out_pa, E_PA);
    edge_msg_kernel<<<(E_AA + 7) / 8, 256, 0, stream>>>(aa_src, aa_dst, xa, exp_aa, denom_aa, out_aa, E_AA);

    sem_score_kernel<<<(N_A + 15) / 16, 256, 0, stream>>>(out_pa, kw, kb, q, score + 0, N_A);
    sem_score_kernel<<<(N_A + 15) / 16, 256, 0, stream>>>(out_aa, kw, kb, q, score + 1, N_A);

    softmax2_kernel<<<1, 32, 0, stream>>>(score, attn);

    fuse_out_kernel<<<(N_A + 15) / 16, 128, 0, stream>>>(out_pa, out_aa, attn, ow, ob,
                                                         logits, Hout, alpha_sem, N_A);
}